// _ProposalLayer_60782377173201
// MI455X (gfx1250) — compile-verified
//
#include <hip/hip_runtime.h>
#include <stdint.h>

#define BATCH     16
#define A_NUM     9
#define HF        160
#define WF        160
#define NANCH     (HF*WF*A_NUM)   /* 230400 */
#define NPAD      262144          /* 2^18 >= NANCH */
#define PRE_TOPK  6000
#define POST_N    300
#define NMS_TH    0.7f
#define MINSZ     16.0f
#define FSTRIDE   16.0f

/* generate_anchors(16, [0.5,1,2], [8,16,32]) precomputed */
__constant__ float c_anchors[A_NUM][4] = {
  { -84.f,  -40.f,  99.f,  55.f},
  {-176.f,  -88.f, 191.f, 103.f},
  {-360.f, -184.f, 375.f, 199.f},
  { -56.f,  -56.f,  71.f,  71.f},
  {-120.f, -120.f, 135.f, 135.f},
  {-248.f, -248.f, 263.f, 263.f},
  { -36.f,  -80.f,  51.f,  95.f},
  { -80.f, -168.f,  95.f, 183.f},
  {-168.f, -344.f, 183.f, 359.f}
};

/* ------------------- CDNA5 async global->LDS path (guarded) ------------- */
#if defined(__has_builtin)
#  if __has_builtin(__builtin_amdgcn_global_load_async_to_lds_b128)
#    define HAVE_ASYNC_LDS 1
#  endif
#  if __has_builtin(__builtin_amdgcn_tensor_load_to_lds)
#    define HAVE_TDM 1
#  endif
#endif

#ifdef HAVE_ASYNC_LDS
typedef int v4i __attribute__((vector_size(16)));
typedef __attribute__((address_space(1))) v4i* g_v4i_ptr;
typedef __attribute__((address_space(3))) v4i* l_v4i_ptr;
__device__ __forceinline__ void async_copy16(const void* gsrc, void* ldst) {
  __builtin_amdgcn_global_load_async_to_lds_b128(
      (g_v4i_ptr)(gsrc), (l_v4i_ptr)(ldst), 0, 0);
}
__device__ __forceinline__ void async_wait0() {
#if __has_builtin(__builtin_amdgcn_s_wait_asynccnt)
  __builtin_amdgcn_s_wait_asynccnt(0);
#else
  asm volatile("s_wait_asynccnt 0" ::: "memory");
#endif
}
#endif

#ifdef HAVE_TDM
typedef unsigned int u32x4 __attribute__((ext_vector_type(4)));
typedef int          i32x8 __attribute__((ext_vector_type(8)));
typedef int          i32x4 __attribute__((ext_vector_type(4)));
typedef __attribute__((address_space(3))) void as3_raw;

/* TDM: linear copy of `nelem` 8-byte elements from global addr -> LDS.
   D# group0: count=1, lds_addr, global_addr[56:0], type=2 ("image").
   D# group1: data_size=3 (8B), tensor_dim0=tile_dim0=nelem,
              tensor_dim1=tile_dim1=1, dim0_stride=nelem.                  */
__device__ __forceinline__ void tdm_load_rows_b64(const void* gsrc, void* ldst,
                                                  unsigned nelem) {
  unsigned long long ga = (unsigned long long)(uintptr_t)gsrc;
  unsigned lds_off = (unsigned)(uintptr_t)(as3_raw*)ldst;
  u32x4 g0;
  g0.x = 1u;                                        /* count=1              */
  g0.y = lds_off;                                   /* lds_addr             */
  g0.z = (unsigned)(ga & 0xFFFFFFFFu);              /* global_addr[31:0]    */
  g0.w = (unsigned)((ga >> 32) & 0x01FFFFFFu) | (2u << 30); /* [56:32]|type */
  i32x8 g1;
  g1[0] = (int)(3u << 16);                          /* data_size = 8 bytes  */
  g1[1] = (int)((nelem & 0xFFFFu) << 16);           /* tensor_dim0[15:0]    */
  g1[2] = (int)(((nelem >> 16) & 0xFFFFu) | (1u << 16)); /* dim0 hi, dim1=1 */
  g1[3] = (int)((nelem & 0xFFFFu) << 16);           /* tile_dim0 (<=65535)  */
  g1[4] = 1;                                        /* tile_dim1=1          */
  g1[5] = (int)nelem;                               /* dim0_stride[31:0]    */
  g1[6] = 0;
  g1[7] = 0;
  i32x4 gz4 = {0, 0, 0, 0};
  i32x8 gz8 = {0, 0, 0, 0, 0, 0, 0, 0};
  __builtin_amdgcn_tensor_load_to_lds(g0, g1, gz4, gz4, gz8, 0);
}
__device__ __forceinline__ void tdm_wait0() {
#if __has_builtin(__builtin_amdgcn_s_wait_tensorcnt)
  __builtin_amdgcn_s_wait_tensorcnt(0);
#else
  asm volatile("s_wait_tensorcnt 0" ::: "memory");
#endif
}
#endif

/* ------------------------- shared box decode ---------------------------- */
__device__ __forceinline__ float4 decode_box(const float* __restrict__ deltas,
                                             float imH, float imW,
                                             int b, int i) {
  int a   = i % A_NUM;
  int pix = i / A_NUM;
  int wx  = pix % WF;
  int hy  = pix / WF;
  float sx = (float)wx * FSTRIDE;
  float sy = (float)hy * FSTRIDE;
  float ax1 = c_anchors[a][0] + sx, ay1 = c_anchors[a][1] + sy;
  float ax2 = c_anchors[a][2] + sx, ay2 = c_anchors[a][3] + sy;
  float aw  = ax2 - ax1 + 1.f,  ah  = ay2 - ay1 + 1.f;
  float acx = ax1 + 0.5f * aw,  acy = ay1 + 0.5f * ah;

  const int plane = HF * WF;
  int dbase = ((b * (4 * A_NUM) + 4 * a) * plane) + hy * WF + wx;
  float d0 = deltas[dbase];
  float d1 = deltas[dbase + plane];
  float d2 = deltas[dbase + 2 * plane];
  float d3 = deltas[dbase + 3 * plane];

  float pcx = d0 * aw + acx;
  float pcy = d1 * ah + acy;
  float pw  = expf(d2) * aw;
  float ph  = expf(d3) * ah;

  float4 r;
  r.x = fminf(fmaxf(pcx - 0.5f * pw, 0.f), imW - 1.f);
  r.y = fminf(fmaxf(pcy - 0.5f * ph, 0.f), imH - 1.f);
  r.z = fminf(fmaxf(pcx + 0.5f * pw, 0.f), imW - 1.f);
  r.w = fminf(fmaxf(pcy + 0.5f * ph, 0.f), imH - 1.f);
  return r;
}

/* -------- kernel 1: decode + filter + pack monotone 64-bit sort key ------ */
__global__ void decode_score_keys(const float* __restrict__ scores_map,
                                  const float* __restrict__ deltas,
                                  const float* __restrict__ im_info,
                                  unsigned long long* __restrict__ keys) {
  int b = blockIdx.y;
  int i = blockIdx.x * blockDim.x + threadIdx.x;
  if (i >= NPAD) return;
  unsigned long long key = 0ull;
  if (i < NANCH) {
    int a   = i % A_NUM;
    int pix = i / A_NUM;
    int wx  = pix % WF;
    int hy  = pix / WF;
    float imH   = im_info[b * 3 + 0];
    float imW   = im_info[b * 3 + 1];
    float scale = im_info[b * 3 + 2];
    float4 bx = decode_box(deltas, imH, imW, b, i);
    float s = scores_map[((b * (2 * A_NUM) + A_NUM + a) * HF + hy) * WF + wx];
    float ms = MINSZ * scale;
    bool keep = ((bx.z - bx.x + 1.f) >= ms) && ((bx.w - bx.y + 1.f) >= ms);
    if (!keep) s = -__builtin_inff();
    unsigned u = __float_as_uint(s);
    u = (u & 0x80000000u) ? ~u : (u | 0x80000000u);     /* monotone total order */
    key = ((unsigned long long)u << 32) | (unsigned)(~(unsigned)i);
  }
  keys[(size_t)b * NPAD + i] = key;
}

/* ---------------- bitonic sort (descending), 2048-chunk LDS kernels ------ */
__device__ __forceinline__ void load_chunk_lds(unsigned long long* __restrict__ g,
                                               unsigned long long* lk, int t) {
#ifdef HAVE_ASYNC_LDS
  async_copy16(g + 2 * t, &lk[2 * t]);   /* 16B/lane = 2 keys, 16KB per block */
  async_wait0();
  __syncthreads();
#else
  lk[t]        = g[t];
  lk[t + 1024] = g[t + 1024];
  __syncthreads();
#endif
}

__global__ __launch_bounds__(1024)
void bitonic_local_sort(unsigned long long* __restrict__ keys) {
  __shared__ unsigned long long lk[2048];
  int b = blockIdx.y;
  int base = blockIdx.x * 2048;               /* within-batch element offset  */
  unsigned long long* g = keys + (size_t)b * NPAD + base;
  int t = threadIdx.x;
  load_chunk_lds(g, lk, t);
  for (int size = 2; size <= 2048; size <<= 1) {
    for (int stride = size >> 1; stride > 0; stride >>= 1) {
      int i = ((t & ~(stride - 1)) << 1) | (t & (stride - 1));
      int l = i + stride;
      bool up = (((base + i) & size) == 0);   /* descending where up==true    */
      unsigned long long a = lk[i], c = lk[l];
      if ((a < c) == up) { lk[i] = c; lk[l] = a; }
      __syncthreads();
    }
  }
  g[t]        = lk[t];
  g[t + 1024] = lk[t + 1024];
}

__global__ __launch_bounds__(1024)
void bitonic_local_merge(unsigned long long* __restrict__ keys, int size) {
  __shared__ unsigned long long lk[2048];
  int b = blockIdx.y;
  int base = blockIdx.x * 2048;
  unsigned long long* g = keys + (size_t)b * NPAD + base;
  int t = threadIdx.x;
  load_chunk_lds(g, lk, t);
  for (int stride = 1024; stride > 0; stride >>= 1) {
    int i = ((t & ~(stride - 1)) << 1) | (t & (stride - 1));
    int l = i + stride;
    bool up = (((base + i) & size) == 0);
    unsigned long long a = lk[i], c = lk[l];
    if ((a < c) == up) { lk[i] = c; lk[l] = a; }
    __syncthreads();
  }
  g[t]        = lk[t];
  g[t + 1024] = lk[t + 1024];
}

__global__ void bitonic_global_step(unsigned long long* __restrict__ keys,
                                    int size, int stride) {
  int b = blockIdx.y;
  int p = blockIdx.x * blockDim.x + threadIdx.x;
  if (p >= NPAD / 2) return;
  int i = ((p & ~(stride - 1)) << 1) | (p & (stride - 1));
  int l = i + stride;
  unsigned long long* g = keys + (size_t)b * NPAD;
  unsigned long long a = g[i], c = g[l];
  bool up = ((i & size) == 0);
  if ((a < c) == up) { g[i] = c; g[l] = a; }
}

/* -------- kernel 3: greedy NMS per batch, candidates resident in LDS -----
   LDS layout (dynamic): sKeys[6000] u64 | sBox[6000] float4 | sElig[6000] u8 */
__global__ __launch_bounds__(1024)
void nms_kernel(const unsigned long long* __restrict__ keys,
                const float* __restrict__ deltas,
                const float* __restrict__ im_info,
                float* __restrict__ out) {
  extern __shared__ unsigned char smem[];
  unsigned long long* sKeys = (unsigned long long*)smem;           /* 48000 B */
  float4*        sBox  = (float4*)(smem + PRE_TOPK * 8);           /* 96000 B */
  unsigned char* sElig = smem + PRE_TOPK * 8 + PRE_TOPK * 16;      /*  6000 B */
  __shared__ int    s_sel;
  __shared__ int    s_ptr;
  __shared__ float4 s_b;

  int b = blockIdx.x;
  int t = threadIdx.x;
  float imH = im_info[b * 3 + 0];
  float imW = im_info[b * 3 + 1];

  /* stage top-PRE_TOPK keys into LDS via the Tensor Data Mover */
  const unsigned long long* gKeys = keys + (size_t)b * NPAD;
#ifdef HAVE_TDM
  if (t < 32) {                 /* exactly one wave issues the DMA           */
    if (t == 0) tdm_load_rows_b64(gKeys, sKeys, PRE_TOPK);
    tdm_wait0();
  }
  __syncthreads();
#else
  for (int r = t; r < PRE_TOPK; r += blockDim.x) sKeys[r] = gKeys[r];
  __syncthreads();
#endif

  /* rank r -> original index, re-decode box */
  for (int r = t; r < PRE_TOPK; r += blockDim.x) {
    unsigned long long k = sKeys[r];
    unsigned hi  = (unsigned)(k >> 32);
    unsigned idx = ~(unsigned)(k & 0xFFFFFFFFull);
    sBox[r]  = decode_box(deltas, imH, imW, b, (int)idx);
    sElig[r] = (hi > 0x007FFFFFu) ? (unsigned char)1 : (unsigned char)0;
  }
  if (t == 0) s_ptr = 0;
  __syncthreads();

  for (int it = 0; it < POST_N; ++it) {
    if (t == 0) {
      int p = s_ptr;
      while (p < PRE_TOPK && !sElig[p]) ++p;     /* sorted => argmax = first */
      float* o = out + ((size_t)b * POST_N + it) * 5;
      o[0] = (float)b;
      if (p < PRE_TOPK) {
        s_sel = p;  s_b = sBox[p];  sElig[p] = 0;  s_ptr = p + 1;
        o[1] = s_b.x; o[2] = s_b.y; o[3] = s_b.z; o[4] = s_b.w;
      } else {
        s_sel = -1; s_ptr = p;
        o[1] = 0.f; o[2] = 0.f; o[3] = 0.f; o[4] = 0.f;
      }
    }
    __syncthreads();
    if (s_sel >= 0) {
      float4 sb = s_b;
      float sbA = (sb.z - sb.x + 1.f) * (sb.w - sb.y + 1.f);
      for (int j = t; j < PRE_TOPK; j += blockDim.x) {
        if (!sElig[j]) continue;
        float4 bj = sBox[j];
        float xx1 = fmaxf(sb.x, bj.x), yy1 = fmaxf(sb.y, bj.y);
        float xx2 = fminf(sb.z, bj.z), yy2 = fminf(sb.w, bj.w);
        float inter = fmaxf(0.f, xx2 - xx1 + 1.f) * fmaxf(0.f, yy2 - yy1 + 1.f);
        float aj  = (bj.z - bj.x + 1.f) * (bj.w - bj.y + 1.f);
        float iou = inter / (sbA + aj - inter);
        if (iou > NMS_TH) sElig[j] = 0;
      }
    }
    __syncthreads();
  }
}

/* ------------------------------- launcher ------------------------------- */
extern "C" void kernel_launch(void* const* d_in, const int* in_sizes, int n_in,
                              void* d_out, int out_size, void* d_ws, size_t ws_size,
                              hipStream_t stream) {
  const float* scores = (const float*)d_in[0];   /* (16, 18, 160, 160) */
  const float* deltas = (const float*)d_in[1];   /* (16, 36, 160, 160) */
  const float* iminfo = (const float*)d_in[2];   /* (16, 3)            */
  float* out = (float*)d_out;                    /* (16, 300, 5)       */
  unsigned long long* keys = (unsigned long long*)d_ws;  /* 16*2^18*8 = 32 MB */

  dim3 gDecode(NPAD / 256, BATCH);
  decode_score_keys<<<gDecode, 256, 0, stream>>>(scores, deltas, iminfo, keys);

  dim3 gLocal(NPAD / 2048, BATCH);
  bitonic_local_sort<<<gLocal, 1024, 0, stream>>>(keys);
  for (int size = 4096; size <= NPAD; size <<= 1) {
    for (int stride = size >> 1; stride >= 2048; stride >>= 1) {
      dim3 gStep((NPAD / 2 + 255) / 256, BATCH);
      bitonic_global_step<<<gStep, 256, 0, stream>>>(keys, size, stride);
    }
    bitonic_local_merge<<<gLocal, 1024, 0, stream>>>(keys, size);
  }

  /* sKeys 48000 + sBox 96000 + sElig 6000 = 150000 B of LDS (<= 320KB/WGP) */
  size_t smem = (size_t)PRE_TOPK * 8 + (size_t)PRE_TOPK * 16 + (size_t)PRE_TOPK;
  nms_kernel<<<BATCH, 1024, smem, stream>>>(keys, deltas, iminfo, out);
}